// SobelEdge3D_24017457119243
// MI455X (gfx1250) — compile-verified
//
#include <hip/hip_runtime.h>
#include <math.h>

// Sobel-3D edge magnitude, fp32, shape (B=2,C=16,D=64,H=128,W=128).
// Memory-bound separable stencil: async global->LDS staging (CDNA5 ASYNCcnt
// path), rolling D-window in a 4-slot LDS ring, separable compute with a
// register cache of the previous two slices' partial reductions.

#define NBC   32            // B*C images
#define DD    64
#define HH    128
#define WW    128
#define HT    8             // output rows per block
#define ROWS  10            // HT + 2 halo rows
#define RST   132           // padded LDS row stride (floats); cols 128..131 stay 0
#define SLOTF (ROWS * RST)  // 1320 floats / slice slot (5280 B, 16B-aligned)
#define NSLOT 4

__device__ __forceinline__ unsigned lds_off_b(const float* p) {
    return (unsigned)(unsigned long long)
        (__attribute__((address_space(3))) const void*)p;
}

// One 16-byte async global->LDS transfer (per active lane).
__device__ __forceinline__ void async_load_vec4(const float* gsrc, const float* ldst) {
    asm volatile("global_load_async_to_lds_b128 %0, %1, off"
                 :: "v"(lds_off_b(ldst)), "v"((unsigned long long)gsrc)
                 : "memory");
}

__device__ __forceinline__ void wait_async0() {
    asm volatile("s_wait_asynccnt 0x0" ::: "memory");
}

__global__ __launch_bounds__(256, 2)
void sobel3d_kernel(const float* __restrict__ in, float* __restrict__ out) {
    __shared__ __align__(16) float sm[NSLOT * SLOTF];

    const int t  = threadIdx.x;
    const int bc = blockIdx.x >> 4;            // 16 h-tiles per image
    const int h0 = (blockIdx.x & 15) * HT;

    // Zero the whole ring once: provides the never-overwritten h/w halo zeros.
    for (int i = t; i < NSLOT * SLOTF; i += 256) sm[i] = 0.0f;
    __syncthreads();

    const size_t img = (size_t)bc * DD * HH * WW;

    // slice dd (global d index) -> slot s ; slot(s) == (s+1)&3
    auto load_slice = [&](int dd, int s) {
        const float* gbase = in + img + (size_t)dd * HH * WW;
        const float* lbase = &sm[s * SLOTF];
#pragma unroll
        for (int rep = 0; rep < 2; ++rep) {
            int v    = t + rep * 256;          // 320 vec4 transfers per slice
            int row  = v >> 5;                 // 0..9
            int col4 = v & 31;                 // 0..31 -> cols 4*col4..4*col4+3
            int hrow = h0 - 1 + row;
            if (v < 320 && hrow >= 0 && hrow < HH) {
                async_load_vec4(gbase + (size_t)hrow * WW + col4 * 4,
                                lbase + row * RST + col4 * 4);
            }
        }
    };
    auto zero_slot = [&](int s) {
        for (int i = t; i < SLOTF; i += 256) sm[s * SLOTF + i] = 0.0f;
    };

    // Thread geometry: 2 half-tiles of 4 output rows x 128 cols.
    const int w     = t & 127;
    const int rbase = (t >> 7) * 4;            // 0 or 4
    const int iL    = (w == 0) ? 129 : (w - 1);  // col 129 is a guaranteed zero
    const int iR    = w + 1;                     // w==127 -> col 128 (zero)

    // Per-slice partial reductions for this thread's 4 output rows:
    //   A = sh(dw)  (-> edge_d term), B = dh(sw) (-> edge_h), C = sh(sw) (-> edge_w)
    auto calcABC = [&](int s, float* A, float* B, float* C) {
        const float* sl = &sm[s * SLOTF + rbase * RST];
        float rs[6], rd[6];
#pragma unroll
        for (int i = 0; i < 6; ++i) {
            const float* rp = sl + i * RST;
            float xl = rp[iL], xc = rp[w], xr = rp[iR];
            rs[i] = xl + 2.0f * xc + xr;       // [1,2,1] along w
            rd[i] = xr - xl;                   // [-1,0,1] along w
        }
#pragma unroll
        for (int k = 0; k < 4; ++k) {
            A[k] = rd[k] + 2.0f * rd[k + 1] + rd[k + 2];   // sh over h
            B[k] = rs[k + 2] - rs[k];                      // dh over h
            C[k] = rs[k] + 2.0f * rs[k + 1] + rs[k + 2];   // sh over h
        }
    };

    float Aa[4] = {0, 0, 0, 0}, Ba[4] = {0, 0, 0, 0}, Ca[4] = {0, 0, 0, 0}; // slice d-1
    float Ab[4], Bb[4], Cb[4];                                              // slice d
    float Ac[4], Bc[4], Cc[4];                                              // slice d+1

    // Prologue: stage slices 0,1,2 into slots 1,2,3; slice -1 is register zeros.
    load_slice(0, 1);
    load_slice(1, 2);
    load_slice(2, 3);
    wait_async0();
    __syncthreads();
    calcABC(1, Ab, Bb, Cb);                    // slice 0

    for (int d = 0; d < DD; ++d) {
        wait_async0();                          // prefetch from iter d-1 done
        __syncthreads();                        // publish; everyone past old reads

        // Prefetch slice d+3 into slot (d+4)&3 == d&3 (held slice d-1, dead).
        if (d <= 60)       load_slice(d + 3, d & 3);
        else if (d == 61)  zero_slot(1);        // slice 64 := zeros (slot(64)==1)

        calcABC((d + 2) & 3, Ac, Bc, Cc);       // slice d+1

        const size_t ob = img + ((size_t)d * HH + (h0 + rbase)) * WW + w;
#pragma unroll
        for (int k = 0; k < 4; ++k) {
            float ed = Aa[k] + 2.0f * Ab[k] + Ac[k];   // sd over d
            float eh = Ba[k] + 2.0f * Bb[k] + Bc[k];   // sd over d
            float ew = Cc[k] - Ca[k];                  // dd over d
            float v  = sqrtf(ed * ed + eh * eh + ew * ew + 1e-6f);
            __builtin_nontemporal_store(v, &out[ob + (size_t)k * WW]);
        }
#pragma unroll
        for (int k = 0; k < 4; ++k) {           // rotate the slice window
            Aa[k] = Ab[k]; Ba[k] = Bb[k]; Ca[k] = Cb[k];
            Ab[k] = Ac[k]; Bb[k] = Bc[k]; Cb[k] = Cc[k];
        }
    }
}

extern "C" void kernel_launch(void* const* d_in, const int* in_sizes, int n_in,
                              void* d_out, int out_size, void* d_ws, size_t ws_size,
                              hipStream_t stream) {
    (void)in_sizes; (void)n_in; (void)d_ws; (void)ws_size; (void)out_size;
    const float* x = (const float*)d_in[0];
    float* y = (float*)d_out;
    dim3 grid(NBC * (HH / HT));   // 32 images * 16 h-tiles = 512 blocks
    dim3 block(256);              // 8 wave32 waves
    sobel3d_kernel<<<grid, block, 0, stream>>>(x, y);
}